// Net_50886772523473
// MI455X (gfx1250) — compile-verified
//
#include <hip/hip_runtime.h>

typedef __attribute__((ext_vector_type(2))) float v2f;
typedef __attribute__((ext_vector_type(8))) float v8f;

#define D 128
#define TSTRIDE 132   // LDS row stride (floats): (r*132+k)%64 unique per r -> no bank conflicts

// ---------------- degree / normalization ----------------
__global__ void k_zero(float* __restrict__ p, int n) {
  int i = blockIdx.x * blockDim.x + threadIdx.x;
  if (i < n) p[i] = 0.0f;
}

__global__ void k_deg(const int* __restrict__ dst, float* __restrict__ deg, int E) {
  int i = blockIdx.x * blockDim.x + threadIdx.x;
  if (i < E) atomicAdd(&deg[dst[i]], 1.0f);
}

__global__ void k_dinv(float* __restrict__ d, int n) {
  int i = blockIdx.x * blockDim.x + threadIdx.x;
  if (i < n) d[i] = rsqrtf(d[i] + 1.0f);   // +1 self-loop; always >= 1
}

// ---------------- H = act(X) @ W via V_WMMA_F32_16X16X4_F32 ----------------
// Block = 256 threads = 8 waves; block handles 16 rows, wave w handles cols [16w,16w+16).
__global__ void k_gemm_wmma(const float* __restrict__ X, const float* __restrict__ W,
                            float* __restrict__ H, int n, int relu) {
  __shared__ float tX[16 * TSTRIDE];
  const int tid = threadIdx.x;
  const int rowBase = blockIdx.x * 16;

  // coalesced stage of the 16x128 A-tile (ReLU of previous layer fused here)
  for (int i = tid; i < 16 * D; i += 256) {
    int r = i >> 7, c = i & (D - 1);
    int row = rowBase + r;
    float v = (row < n) ? X[(size_t)row * D + c] : 0.0f;
    if (relu) v = fmaxf(v, 0.0f);
    tX[r * TSTRIDE + c] = v;
  }
  __syncthreads();

  const int wave    = tid >> 5;
  const int lane    = tid & 31;
  const int colBase = wave * 16;
  const int r       = lane & 15;   // A: row in tile / B,C,D: column in tile
  const int half    = lane >> 4;

  v8f acc = {};
  #pragma unroll
  for (int k0 = 0; k0 < D; k0 += 4) {
    const int ka = k0 + half * 2;
    v2f a, b;
    // A 16x4 f32 fragment: lanes 0-15 K={k0,k0+1}, lanes 16-31 K={k0+2,k0+3}
    a.x = tX[r * TSTRIDE + ka];
    a.y = tX[r * TSTRIDE + ka + 1];
    // B 4x16 f32 fragment (row-major W[128][128]): same K split, N = r
    b.x = W[ka * D + colBase + r];
    b.y = W[(ka + 1) * D + colBase + r];
    acc = __builtin_amdgcn_wmma_f32_16x16x4_f32(false, a, false, b, (short)0, acc,
                                                false, false);
  }

  // C/D layout: VGPR i -> M = i + 8*half, N = lane&15
  const int outRow0 = rowBase + half * 8;
  #pragma unroll
  for (int i = 0; i < 8; ++i) {
    int row = outRow0 + i;
    if (row < n) H[(size_t)row * D + colBase + r] = acc[i];
  }
}

// ---------------- out[i] = bias + dinv[i]^2 * h[i]  (self-loop term) ----------------
__global__ void k_init(const float* __restrict__ h, const float* __restrict__ dinv,
                       const float* __restrict__ bias, float* __restrict__ out, int n) {
  int i = blockIdx.x * blockDim.x + threadIdx.x;
  if (i < n * D) {
    int node = i >> 7, f = i & (D - 1);
    float di = dinv[node];
    out[i] = bias[f] + di * di * h[i];
  }
}

// ---------------- edge scatter: one wave32 per edge, 128 f32 atomics into L2 ----------------
__global__ void k_scatter(const int* __restrict__ src, const int* __restrict__ dst,
                          const float* __restrict__ dinv, const float* __restrict__ h,
                          float* __restrict__ out, int E) {
  int wid  = (blockIdx.x * blockDim.x + threadIdx.x) >> 5;
  int lane = threadIdx.x & 31;
  if (wid >= E) return;
  int s = src[wid], d = dst[wid];          // wave-uniform -> scalarized
  float nrm = dinv[s] * dinv[d];
  const float4 m = ((const float4*)(h + (size_t)s * D))[lane];
  float* op = out + (size_t)d * D + lane * 4;
  atomicAdd(op + 0, nrm * m.x);
  atomicAdd(op + 1, nrm * m.y);
  atomicAdd(op + 2, nrm * m.z);
  atomicAdd(op + 3, nrm * m.w);
}

// ---------------- decoder: one wave32 per scored edge ----------------
__global__ void k_decode(const int* __restrict__ u, const int* __restrict__ v,
                         const float* __restrict__ x, float* __restrict__ out, int EL) {
  int wid  = (blockIdx.x * blockDim.x + threadIdx.x) >> 5;
  int lane = threadIdx.x & 31;
  if (wid >= EL) return;
  int a = u[wid], b = v[wid];
  const float4 p = ((const float4*)(x + (size_t)a * D))[lane];
  const float4 q = ((const float4*)(x + (size_t)b * D))[lane];
  float s = p.x * q.x + p.y * q.y + p.z * q.z + p.w * q.w;
  #pragma unroll
  for (int off = 16; off > 0; off >>= 1) s += __shfl_xor(s, off, 32);
  if (lane == 0) out[wid] = s;
}

// ---------------- orchestration ----------------
extern "C" void kernel_launch(void* const* d_in, const int* in_sizes, int n_in,
                              void* d_out, int out_size, void* d_ws, size_t ws_size,
                              hipStream_t stream) {
  (void)n_in; (void)out_size; (void)ws_size;
  const float* X0 = (const float*)d_in[0];
  const float* W1 = (const float*)d_in[1];
  const float* b1 = (const float*)d_in[2];
  const float* W2 = (const float*)d_in[3];
  const float* b2 = (const float*)d_in[4];
  const float* W3 = (const float*)d_in[5];
  const float* b3 = (const float*)d_in[6];
  const int* ei   = (const int*)d_in[7];   // [2, E]
  const int* eli  = (const int*)d_in[8];   // [2, EL]
  float* out      = (float*)d_out;

  const int N  = in_sizes[0] / D;
  const int E  = in_sizes[7] / 2;
  const int EL = in_sizes[8] / 2;
  const int* src = ei;
  const int* dst = ei + E;
  const int* lu  = eli;
  const int* lv  = eli + EL;

  // workspace: dinv[N] | h[N*D] | xA[N*D]   (~51.4 MB)
  char* ws = (char*)d_ws;
  float* dinv = (float*)ws;
  size_t off = ((size_t)N * 4 + 255) & ~(size_t)255;
  float* h = (float*)(ws + off);
  off += ((size_t)N * D * 4 + 255) & ~(size_t)255;
  float* xA = (float*)(ws + off);

  dim3 blk(256);
  const int gN   = (N + 255) / 256;
  const int gE   = (E + 255) / 256;
  const int gG   = (N + 15) / 16;                         // GEMM row strips
  const int gND  = (N * D + 255) / 256;
  const int gEd  = (int)(((size_t)E * 32 + 255) / 256);   // 1 wave per edge
  const int gEL  = (int)(((size_t)EL * 32 + 255) / 256);  // 1 wave per scored edge

  // symmetric normalization
  k_zero<<<gN, blk, 0, stream>>>(dinv, N);
  k_deg <<<gE, blk, 0, stream>>>(dst, dinv, E);
  k_dinv<<<gN, blk, 0, stream>>>(dinv, N);

  // layer 1
  k_gemm_wmma<<<gG,  blk, 0, stream>>>(X0, W1, h, N, 0);
  k_init     <<<gND, blk, 0, stream>>>(h, dinv, b1, xA, N);
  k_scatter  <<<gEd, blk, 0, stream>>>(src, dst, dinv, h, xA, E);
  // layer 2 (ReLU of layer-1 output fused into A-tile load)
  k_gemm_wmma<<<gG,  blk, 0, stream>>>(xA, W2, h, N, 1);
  k_init     <<<gND, blk, 0, stream>>>(h, dinv, b2, xA, N);
  k_scatter  <<<gEd, blk, 0, stream>>>(src, dst, dinv, h, xA, E);
  // layer 3 (ReLU of layer-2 output fused; no ReLU on output)
  k_gemm_wmma<<<gG,  blk, 0, stream>>>(xA, W3, h, N, 1);
  k_init     <<<gND, blk, 0, stream>>>(h, dinv, b3, xA, N);
  k_scatter  <<<gEd, blk, 0, stream>>>(src, dst, dinv, h, xA, E);

  // dot-product edge decoder
  k_decode<<<gEL, blk, 0, stream>>>(lu, lv, xA, out, EL);
}